// DilatedSlidingWindowAttention_27685359190793
// MI455X (gfx1250) — compile-verified
//
#include <hip/hip_runtime.h>
#include <math.h>

// ---------------- constants -----------------
#define SEQ   2000
#define DM    768
#define NH    12
#define HD    64
#define WIN   16
#define DIL   2
#define NB    2
#define NGl   50          // global tokens per batch
#define NLc   1950        // local tokens per batch
#define KROWS 1980        // NL + (WIN-1)*DIL padded rows
#define WINXD 12288       // WIN * DM
#define CHUNK 1024        // WIN * HD

typedef float v2f __attribute__((ext_vector_type(2)));
typedef float v4f __attribute__((ext_vector_type(4)));
typedef float v8f __attribute__((ext_vector_type(8)));

// src-row mapping modes
#define SRC_DIRECT    0
#define SRC_LOCAL     1
#define SRC_LOCAL_PAD 2
#define SRC_GLOBAL    3
// dst-row mapping modes
#define DST_DIRECT 0
#define DST_LOCAL  1
#define DST_GLOBAL 2

__device__ __forceinline__ int local_n_to_t(int n) {
    // LOCAL_IDX = all t in [0,2000) with t % 40 != 0, in order
    return (n / 39) * 40 + (n % 39) + 1;
}

__device__ __forceinline__ int map_src_row(int r, int mode, int rowsPerB) {
    if (mode == SRC_DIRECT) return r;
    int b = r / rowsPerB;
    int n = r - b * rowsPerB;
    if (mode == SRC_LOCAL)      return b * SEQ + local_n_to_t(n);
    if (mode == SRC_LOCAL_PAD)  return (n >= NLc) ? -1 : (b * SEQ + local_n_to_t(n));
    /* SRC_GLOBAL */            return b * SEQ + n * 40;
}

__device__ __forceinline__ int map_dst_row(int r, int mode, int rowsPerB) {
    if (mode == DST_DIRECT) return r;
    int b = r / rowsPerB;
    int n = r - b * rowsPerB;
    int t = (mode == DST_LOCAL) ? local_n_to_t(n) : (n * 40);
    return b * SEQ + t;
}

// ---------------- small utility ----------------
__global__ void zero_fill_kernel(float* __restrict__ p, int n) {
    int i = blockIdx.x * 256 + threadIdx.x;
    if (i < n) p[i] = 0.0f;
}

// ---------------- weight transpose (Wt[c][j] = W[j][c]) ----------------
__global__ void transpose768_kernel(const float* __restrict__ in, float* __restrict__ out) {
    __shared__ float tile[16][17];
    int bx = blockIdx.x * 16, by = blockIdx.y * 16;
    int tx = threadIdx.x, ty = threadIdx.y;
    tile[ty][tx] = in[(size_t)(by + ty) * DM + bx + tx];
    __syncthreads();
    out[(size_t)(bx + ty) * DM + by + tx] = tile[tx][ty];
}

// ---------------- fp32 WMMA GEMM: Out = gather(X) @ Wt + bias, scatter rows ----
// block = 128 threads (4 waves); block tile = 64 rows x 64 cols; wave tile = 16x64
__global__ void wmma_gemm_f32_kernel(const float* __restrict__ X,
                                     const float* __restrict__ Wt,   // [DM][DM], Wt[c][j] = W[j][c]
                                     const float* __restrict__ bias, // [DM]
                                     const float* __restrict__ zrow, // 768 zeros
                                     float* __restrict__ Out,
                                     int M, int srcMode, int srcRowsPerB,
                                     int dstMode, int dstRowsPerB) {
    __shared__ __align__(16) float lB[16][72];  // 16 K x 64 N chunk; stride 72 kills bank conflicts

    const int lane = threadIdx.x & 31;
    const int wave = threadIdx.x >> 5;
    const int row0 = blockIdx.x * 64 + wave * 16;
    const int col0 = blockIdx.y * 64;

    // per-lane A source row (A 16x4 f32 layout: lanes 0-15 -> M 0-15 K{0,1}; lanes 16-31 -> K{2,3})
    const int arow = row0 + (lane & 15);
    const int klo  = (lane >> 4) * 2;
    int srow = (arow < M) ? map_src_row(arow, srcMode, srcRowsPerB) : -1;
    const float* aptr = (srow >= 0) ? (X + (size_t)srow * DM) : zrow;  // no divergence in loop

    // static B-staging assignment: 2 aligned float4 chunks per thread
    const int ch0 = threadIdx.x * 2;
    const int rr0 = ch0 >> 4, cs0 = (ch0 & 15) * 4;
    const int ch1 = ch0 + 1;
    const int rr1 = ch1 >> 4, cs1 = (ch1 & 15) * 4;

    v8f acc0 = {}, acc1 = {}, acc2 = {}, acc3 = {};
    const int jc = lane & 15;

    for (int c0 = 0; c0 < DM; c0 += 16) {
        __syncthreads();
        *(v4f*)&lB[rr0][cs0] = *(const v4f*)&Wt[(size_t)(c0 + rr0) * DM + col0 + cs0];
        *(v4f*)&lB[rr1][cs1] = *(const v4f*)&Wt[(size_t)(c0 + rr1) * DM + col0 + cs1];
        __syncthreads();

#pragma unroll
        for (int kk = 0; kk < 16; kk += 4) {
            v2f a = *(const v2f*)&aptr[c0 + kk + klo];
            v2f b0, b1, b2, b3;
            b0.x = lB[kk + klo][ 0 + jc]; b0.y = lB[kk + klo + 1][ 0 + jc];
            b1.x = lB[kk + klo][16 + jc]; b1.y = lB[kk + klo + 1][16 + jc];
            b2.x = lB[kk + klo][32 + jc]; b2.y = lB[kk + klo + 1][32 + jc];
            b3.x = lB[kk + klo][48 + jc]; b3.y = lB[kk + klo + 1][48 + jc];
            acc0 = __builtin_amdgcn_wmma_f32_16x16x4_f32(false, a, false, b0, (short)0, acc0, false, false);
            acc1 = __builtin_amdgcn_wmma_f32_16x16x4_f32(false, a, false, b1, (short)0, acc1, false, false);
            acc2 = __builtin_amdgcn_wmma_f32_16x16x4_f32(false, a, false, b2, (short)0, acc2, false, false);
            acc3 = __builtin_amdgcn_wmma_f32_16x16x4_f32(false, a, false, b3, (short)0, acc3, false, false);
        }
    }

    // C/D layout: VGPR v, lanes 0-15 -> M=v, lanes 16-31 -> M=v+8; N = lane&15
    const int rbase = row0 + 8 * (lane >> 4);
#pragma unroll
    for (int t = 0; t < 4; ++t) {
        v8f acc = (t == 0) ? acc0 : (t == 1) ? acc1 : (t == 2) ? acc2 : acc3;
        int col = col0 + t * 16 + jc;
        float bv = bias[col];
#pragma unroll
        for (int v = 0; v < 8; ++v) {
            int r = rbase + v;
            if (r < M) {
                int dr = map_dst_row(r, dstMode, dstRowsPerB);
                Out[(size_t)dr * DM + col] = acc[v] + bv;
            }
        }
    }
}

// ---------------- local attention (head-softmax variant, flat-reinterp gathers) ----
// one block (256 threads) per (b, n); Kfull/Vfull are [B][KROWS][DM]
__global__ void local_attn_kernel(const float* __restrict__ Qp,
                                  const float* __restrict__ Kfull,
                                  const float* __restrict__ Vfull,
                                  float* __restrict__ Lmat) {
    __shared__ __align__(16) float sq[DM];
    __shared__ float sS[NH][WIN];
    __shared__ float sA[NH][WIN];
    __shared__ float colMax[WIN];
    __shared__ float colSum[WIN];

    const int b = blockIdx.x / NLc;
    const int n = blockIdx.x - b * NLc;
    const int tid = threadIdx.x;

    const float* qrow = Qp + (size_t)(b * NLc + n) * DM;
    if (tid < DM / 4) ((v4f*)sq)[tid] = ((const v4f*)qrow)[tid];
    __syncthreads();

    const float* Kb = Kfull + (size_t)b * KROWS * DM;
    const float* Vb = Vfull + (size_t)b * KROWS * DM;

    if (tid < NH * WIN) {
        int h = tid >> 4;
        int w = tid & 15;
        // virtual flat index into (NL, WIN, DM) reinterpreted as (H, NL, WIN, HD):
        // F -> (n0 = F/12288, w0 = (F%12288)/768, c0 = F%768); value = Kb[n0 + 2*w0][c0]
        unsigned F = (unsigned)(h * NLc + n) * CHUNK + (unsigned)w * HD;
        unsigned n0 = F / WINXD;
        unsigned rem = F - n0 * WINXD;
        unsigned w0 = rem / DM;
        unsigned c0 = rem - w0 * DM;
        const float* kp = Kb + (size_t)(n0 + DIL * w0) * DM + c0;
        float s = 0.0f;
        for (int d = 0; d < HD; ++d) {
            s += sq[h * HD + d] * (*kp);
            ++kp;                                   // F += 1 => c0 += 1
            if (++c0 == DM) { c0 = 0; ++w0; kp += DM; }          // row advances by DIL
            if (w0 == WIN)  { w0 = 0; kp -= (2 * WIN - 1) * DM; } // back to row n0+1
        }
        sS[h][w] = s * 0.125f;   // 1/sqrt(64)
    }
    __syncthreads();

    if (tid < WIN) {
        int w = tid;
        float m = sS[0][w];
        for (int h = 1; h < NH; ++h) m = fmaxf(m, sS[h][w]);
        float sum = 0.0f;
        for (int h = 0; h < NH; ++h) sum += expf(sS[h][w] - m);
        colMax[w] = m;
        colSum[w] = sum;
    }
    __syncthreads();

    // softmax over the HEAD axis (matches reference's axis=1 softmax)
    if (tid < NH * WIN) {
        int h = tid >> 4;
        int w = tid & 15;
        sA[h][w] = expf(sS[h][w] - colMax[w]) / colSum[w];
    }
    __syncthreads();

    for (int e = tid; e < DM; e += 256) {
        int h = e >> 6;
        int d = e & 63;
        unsigned F = (unsigned)(h * NLc + n) * CHUNK + (unsigned)d;
        unsigned n0 = F / WINXD;
        unsigned rem = F - n0 * WINXD;
        unsigned w0 = rem / DM;
        unsigned c0 = rem - w0 * DM;
        const float* vp = Vb + (size_t)(n0 + DIL * w0) * DM + c0;
        float o = 0.0f;
#pragma unroll
        for (int w = 0; w < WIN; ++w) {
            o += sA[h][w] * (*vp);
            vp += HD; c0 += HD;                      // F += 64
            if (c0 >= DM) { c0 -= DM; ++w0; vp += DM; }
            if (w0 >= WIN) { w0 = 0; vp -= (2 * WIN - 1) * DM; }
        }
        // output flat reinterpretation: (B,H,NL,HD) -> (B,NL,D)
        Lmat[(size_t)b * NLc * DM + (size_t)h * NLc * HD + (size_t)n * HD + d] = o;
    }
}

// ---------------- global attention (attends across batch axis) ----------------
// one block (64 threads) per (n, h)
__global__ void global_attn_kernel(const float* __restrict__ Qg,
                                   const float* __restrict__ Kg,
                                   const float* __restrict__ Vg,
                                   float* __restrict__ Gmat) {
    __shared__ float sc[NB][NB];
    __shared__ float p[NB][NB];
    const int n = blockIdx.x / NH;
    const int h = blockIdx.x - n * NH;
    const int tid = threadIdx.x;

    if (tid < NB * NB) {
        int l = tid >> 1, s = tid & 1;
        const float* q = Qg + (size_t)(l * NGl + n) * DM + h * HD;
        const float* k = Kg + (size_t)(s * NGl + n) * DM + h * HD;
        float acc = 0.0f;
        for (int d = 0; d < HD; ++d) acc += q[d] * k[d];
        sc[l][s] = acc * 0.125f;
    }
    __syncthreads();
    if (tid < NB) {
        int l = tid;
        float m = fmaxf(sc[l][0], sc[l][1]);
        float e0 = expf(sc[l][0] - m), e1 = expf(sc[l][1] - m);
        float inv = 1.0f / (e0 + e1);
        p[l][0] = e0 * inv;
        p[l][1] = e1 * inv;
    }
    __syncthreads();
    int d = tid;  // 0..63
    float v0 = Vg[(size_t)(0 * NGl + n) * DM + h * HD + d];
    float v1 = Vg[(size_t)(1 * NGl + n) * DM + h * HD + d];
#pragma unroll
    for (int l = 0; l < NB; ++l) {
        Gmat[(size_t)(l * NGl + n) * DM + h * HD + d] = p[l][0] * v0 + p[l][1] * v1;
    }
}

// ---------------- launcher ----------------
extern "C" void kernel_launch(void* const* d_in, const int* in_sizes, int n_in,
                              void* d_out, int out_size, void* d_ws, size_t ws_size,
                              hipStream_t stream) {
    const float* query = (const float*)d_in[0];
    const float* key   = (const float*)d_in[1];
    const float* value = (const float*)d_in[2];
    const float* Wm[8];
    const float* bv[8];
    for (int i = 0; i < 8; ++i) Wm[i] = (const float*)d_in[3 + i];
    for (int i = 0; i < 8; ++i) bv[i] = (const float*)d_in[11 + i];
    // Wm/bv order: 0=q_l 1=k_l 2=v_l 3=o_l 4=q_g 5=k_g 6=v_g 7=o_g
    float* out = (float*)d_out;
    float* ws  = (float*)d_ws;

    size_t off = 0;
    float* Wt[8];
    for (int i = 0; i < 8; ++i) { Wt[i] = ws + off; off += (size_t)DM * DM; }
    float* Qp    = ws + off; off += (size_t)NB * NLc   * DM;
    float* Kfull = ws + off; off += (size_t)NB * KROWS * DM;
    float* Vfull = ws + off; off += (size_t)NB * KROWS * DM;
    float* Lmat  = ws + off; off += (size_t)NB * NLc   * DM;
    float* Qg    = ws + off; off += (size_t)NB * NGl   * DM;
    float* Kg    = ws + off; off += (size_t)NB * NGl   * DM;
    float* Vg    = ws + off; off += (size_t)NB * NGl   * DM;
    float* Gmat  = ws + off; off += (size_t)NB * NGl   * DM;
    float* zrow  = ws + off; off += (size_t)DM;

    // 0) zero row for padded/tail GEMM rows
    zero_fill_kernel<<<(DM + 255) / 256, 256, 0, stream>>>(zrow, DM);

    // 1) transpose all 8 weight matrices
    {
        dim3 tb(16, 16), tg(DM / 16, DM / 16);
        for (int i = 0; i < 8; ++i)
            transpose768_kernel<<<tg, tb, 0, stream>>>(Wm[i], Wt[i]);
    }

    const int NCOL = DM / 64;  // 12 column tiles
    // 2) projections (WMMA f32 GEMMs)
    {
        int M = NB * NLc;      // 3900
        wmma_gemm_f32_kernel<<<dim3((M + 63) / 64, NCOL), 128, 0, stream>>>(
            query, Wt[0], bv[0], zrow, Qp, M, SRC_LOCAL, NLc, DST_DIRECT, 0);
    }
    {
        int M = NB * KROWS;    // 3960
        wmma_gemm_f32_kernel<<<dim3((M + 63) / 64, NCOL), 128, 0, stream>>>(
            key, Wt[1], bv[1], zrow, Kfull, M, SRC_LOCAL_PAD, KROWS, DST_DIRECT, 0);
        wmma_gemm_f32_kernel<<<dim3((M + 63) / 64, NCOL), 128, 0, stream>>>(
            value, Wt[2], bv[2], zrow, Vfull, M, SRC_LOCAL_PAD, KROWS, DST_DIRECT, 0);
    }
    {
        int M = NB * NGl;      // 100
        wmma_gemm_f32_kernel<<<dim3((M + 63) / 64, NCOL), 128, 0, stream>>>(
            query, Wt[4], bv[4], zrow, Qg, M, SRC_GLOBAL, NGl, DST_DIRECT, 0);
        wmma_gemm_f32_kernel<<<dim3((M + 63) / 64, NCOL), 128, 0, stream>>>(
            key, Wt[5], bv[5], zrow, Kg, M, SRC_GLOBAL, NGl, DST_DIRECT, 0);
        wmma_gemm_f32_kernel<<<dim3((M + 63) / 64, NCOL), 128, 0, stream>>>(
            value, Wt[6], bv[6], zrow, Vg, M, SRC_GLOBAL, NGl, DST_DIRECT, 0);
    }

    // 3) attention cores
    local_attn_kernel<<<NB * NLc, 256, 0, stream>>>(Qp, Kfull, Vfull, Lmat);
    global_attn_kernel<<<NGl * NH, 64, 0, stream>>>(Qg, Kg, Vg, Gmat);

    // 4) output projections, scattered straight into d_out (local+global covers all rows)
    {
        int M = NB * NLc;
        wmma_gemm_f32_kernel<<<dim3((M + 63) / 64, NCOL), 128, 0, stream>>>(
            Lmat, Wt[3], bv[3], zrow, out, M, SRC_DIRECT, NLc, DST_LOCAL, NLc);
    }
    {
        int M = NB * NGl;
        wmma_gemm_f32_kernel<<<dim3((M + 63) / 64, NCOL), 128, 0, stream>>>(
            Gmat, Wt[7], bv[7], zrow, out, M, SRC_DIRECT, NGl, DST_GLOBAL, NGl);
    }
}